// S3FVIN_80771154968999
// MI455X (gfx1250) — compile-verified
//
#include <hip/hip_runtime.h>
#include <hip/hip_bf16.h>
#include <math.h>

typedef __attribute__((ext_vector_type(2))) float v2f;
typedef __attribute__((ext_vector_type(8))) float v8f;

#define HS   0.01f
#define EPSP 0.1f

// ---------------- WMMA: D(16x16 f32) = A(16x4 f32) x B(4x16 f32) + 0 -------
__device__ __forceinline__ v8f wmma_k4(v2f a, v2f b) {
  v8f c = {};
  return __builtin_amdgcn_wmma_f32_16x16x4_f32(false, a, false, b, (short)0, c,
                                               false, false);
}

// ---------------- fast scalar math -----------------------------------------
__device__ __forceinline__ float rcpf(float x) { return __builtin_amdgcn_rcpf(x); }

// branchless tanh: 1 - 2/(e^{2|x|}+1), native exp + rcp; saturates correctly
__device__ __forceinline__ float fast_tanh(float x) {
  float ax = fabsf(x);
  float e  = __expf(2.0f * ax);
  float t  = 1.0f - 2.0f * rcpf(e + 1.0f);
  return copysignf(t, x);
}

// symmetric 3x3 stored [00,01,02,11,12,22]
__device__ __forceinline__ void symmv(const float M[6], const float v[3], float r[3]) {
  r[0] = M[0]*v[0] + M[1]*v[1] + M[2]*v[2];
  r[1] = M[1]*v[0] + M[3]*v[1] + M[4]*v[2];
  r[2] = M[2]*v[0] + M[4]*v[1] + M[5]*v[2];
}

__device__ __forceinline__ void inv3sym(const float M[6], float R[6]) {
  float a=M[0],b=M[1],c=M[2],d=M[3],e=M[4],f=M[5];
  float c00 = d*f - e*e, c01 = e*c - b*f, c02 = b*e - d*c;
  float id  = rcpf(a*c00 + b*c01 + c*c02);
  R[0]=c00*id; R[1]=c01*id; R[2]=c02*id;
  R[3]=(a*f - c*c)*id; R[4]=(b*c - a*e)*id; R[5]=(a*d - b*b)*id;
}

// r = H_mat(q) @ d4,  H = [-v | q0*I - skew(v)]
__device__ __forceinline__ void hmat_mul(const float q[4], const float d[4], float r[3]) {
  float q0=q[0], v0=q[1], v1=q[2], v2=q[3];
  r[0] = -v0*d[0] + q0*d[1] + v2*d[2] - v1*d[3];
  r[1] = -v1*d[0] - v2*d[1] + q0*d[2] + v0*d[3];
  r[2] = -v2*d[0] + v1*d[1] - v0*d[2] + q0*d[3];
}

__device__ __forceinline__ void qmul(const float a[4], const float b[4], float r[4]) {
  r[0] = a[0]*b[0] - a[1]*b[1] - a[2]*b[2] - a[3]*b[3];
  r[1] = a[0]*b[1] + a[1]*b[0] + a[2]*b[3] - a[3]*b[2];
  r[2] = a[0]*b[2] - a[1]*b[3] + a[2]*b[0] + a[3]*b[1];
  r[3] = a[0]*b[3] + a[1]*b[2] - a[2]*b[1] + a[3]*b[0];
}

// res(xi) = G(quat_exp(-xi)) @ (Jq @ quat_exp(-xi).vec), plus 3x3 fwd-mode jac
__device__ __forceinline__ void res_and_jac(const float xi[3], const float Jq[6],
                                            float r[3], float jc[9]) {
  float dot = xi[0]*xi[0] + xi[1]*xi[1] + xi[2]*xi[2];
  bool clamped = dot < 1e-24f;
  float t2 = clamped ? 1e-24f : dot;
  float th = sqrtf(t2);
  float ith = rcpf(th);
  float cs = __cosf(th), sn = __sinf(th);
  float so = sn * ith;
  float qv[3] = { -xi[0]*so, -xi[1]*so, -xi[2]*so };
  float q0 = cs;
  float m[3]; symmv(Jq, qv, m);
  r[0] = q0*m[0] - (qv[1]*m[2] - qv[2]*m[1]);
  r[1] = q0*m[1] - (qv[2]*m[0] - qv[0]*m[2]);
  r[2] = q0*m[2] - (qv[0]*m[1] - qv[1]*m[0]);
  float dso_dth = (cs*th - sn) * (ith * ith);
#pragma unroll
  for (int j = 0; j < 3; ++j) {
    float dth  = clamped ? 0.0f : xi[j] * ith;
    float dso  = dso_dth * dth;
    float dq0  = -sn * dth;
    float dqv[3] = { -xi[0]*dso, -xi[1]*dso, -xi[2]*dso };
    dqv[j] -= so;
    float dm[3]; symmv(Jq, dqv, dm);
    jc[0*3+j] = dq0*m[0] + q0*dm[0] - (dqv[1]*m[2]-dqv[2]*m[1]) - (qv[1]*dm[2]-qv[2]*dm[1]);
    jc[1*3+j] = dq0*m[1] + q0*dm[1] - (dqv[2]*m[0]-dqv[0]*m[2]) - (qv[2]*dm[0]-qv[0]*dm[2]);
    jc[2*3+j] = dq0*m[2] + q0*dm[2] - (dqv[0]*m[1]-dqv[1]*m[0]) - (qv[0]*dm[1]-qv[1]*dm[0]);
  }
}

__device__ __forceinline__ void solve3(const float j[9], const float r[3], float x[3]) {
  float a=j[0],b=j[1],c=j[2],d=j[3],e=j[4],f=j[5],g=j[6],h=j[7],i=j[8];
  float A = e*i - f*h, B = f*g - d*i, C = d*h - e*g;
  float id = rcpf(a*A + b*B + c*C);
  x[0] = (r[0]*A + b*(f*r[2]-r[1]*i) + c*(r[1]*h-e*r[2])) * id;
  x[1] = (a*(r[1]*i-f*r[2]) + r[0]*B + c*(d*r[2]-r[1]*g)) * id;
  x[2] = (a*(e*r[2]-r[1]*h) + b*(r[1]*g-d*r[2]) + r[0]*C) * id;
}

// ---------------- WMMA operand builders ------------------------------------
// B(4x16) = W1^T zero-padded: lane L holds column n = L&15; VGPR v holds row
// k = v + 2*(L>=16) (mirrors the documented f32 16x4 A layout K-split).
// Clamped unconditional load + select: no EXEC divergence.
__device__ __forceinline__ v2f make_B(const float* __restrict__ w1, int lane) {
  int hi = lane >> 4, n = lane & 15;
  int nc = (n < 10) ? n : 0;
  float b0 = w1[nc*4 + 2*hi];
  float b1 = w1[nc*4 + 2*hi + 1];
  v2f b;
  b[0] = (n < 10) ? b0 : 0.0f;
  b[1] = (n < 10) ? b1 : 0.0f;
  return b;
}

// One layer-1 GEMM for a wave's 32 elements: 2 WMMAs + intra-wave LDS
// transpose (DS ops are in-order within a wave; no block barrier needed).
// z = q @ W1^T per lane; both tanh branches derived from this single z since
// z(-q) = -z(q).
__device__ __forceinline__ void net_z(v2f A0, v2f A1, v2f B,
                                      float* stage, int lane, float z[10]) {
  v8f d0 = wmma_k4(A0, B);
  v8f d1 = wmma_k4(A1, B);
  const int hi = lane >> 4, col = lane & 15;
#pragma unroll
  for (int v = 0; v < 8; ++v) {
    stage[(8*hi + v)      * 17 + col] = d0[v];
    stage[(16 + 8*hi + v) * 17 + col] = d1[v];
  }
  __builtin_amdgcn_wave_barrier();    // scheduling fence only; DS is in-order
#pragma unroll
  for (int k = 0; k < 10; ++k) z[k] = stage[lane * 17 + k];
  __builtin_amdgcn_wave_barrier();
}

// psd entries from layer-2 output o[6]: P = L L^T + EPS*I (sym, packed)
__device__ __forceinline__ void psd_pack(const float o[6], float P[6]) {
  P[0] = o[0]*o[0] + EPSP;
  P[1] = o[0]*o[3];
  P[2] = o[0]*o[4];
  P[3] = o[3]*o[3] + o[1]*o[1] + EPSP;
  P[4] = o[3]*o[4] + o[1]*o[5];
  P[5] = o[4]*o[4] + o[5]*o[5] + o[2]*o[2] + EPSP;
}

// ---------------- kernel ----------------------------------------------------
__global__ __launch_bounds__(256)
void qvi_kernel(const float* __restrict__ x,
                const float* __restrict__ Jw1, const float* __restrict__ Jb1,
                const float* __restrict__ Jw2, const float* __restrict__ Jb2,
                const float* __restrict__ Vw1, const float* __restrict__ Vb1,
                const float* __restrict__ Vw2,
                const float* __restrict__ Gw1, const float* __restrict__ Gb1,
                const float* __restrict__ Gw2, const float* __restrict__ Gb2,
                float* __restrict__ out) {
  __shared__ float stg[8][32][17];   // per-wave 32x16 transpose tile (padded)
  __shared__ float qst[8][32][4];    // per-wave quaternion staging for A build

  const int lane = threadIdx.x & 31;
  const int wid  = threadIdx.x >> 5;
  const size_t e = (size_t)blockIdx.x * 256 + threadIdx.x;  // element == lane
  float* stage = &stg[wid][0][0];

  // -------- load own element (one pass, 2x B128) --------
  const float4* xe = reinterpret_cast<const float4*>(x) + e * 2;
  float4 lo = xe[0], hi4 = xe[1];
  float q[4]  = { lo.x, lo.y, lo.z, lo.w };
  float wk[3] = { hi4.x, hi4.y, hi4.z };
  float u     = hi4.w;

  // -------- build A tiles for q_k (documented 16x4 f32 A layout) --------
  qst[wid][lane][0]=q[0]; qst[wid][lane][1]=q[1];
  qst[wid][lane][2]=q[2]; qst[wid][lane][3]=q[3];
  __builtin_amdgcn_wave_barrier();
  const int half = lane >> 4, cm = lane & 15;
  v2f A0, A1;
  A0[0] = qst[wid][cm][2*half];      A0[1] = qst[wid][cm][2*half + 1];
  A1[0] = qst[wid][16+cm][2*half];   A1[1] = qst[wid][16+cm][2*half + 1];
  __builtin_amdgcn_wave_barrier();

  const v2f BJ = make_B(Jw1, lane);
  const v2f BV = make_B(Vw1, lane);
  const v2f BG = make_B(Gw1, lane);

  float z[10], t[10], o[6];

  // -------- J net: J_q_inv = 0.5*(psd(q)+psd(-q)); z(-q) = -z(q) --------
  float Ji[6], Pm[6];
  net_z(A0, A1, BJ, stage, lane, z);
#pragma unroll
  for (int k = 0; k < 10; ++k) t[k] = fast_tanh(z[k] + Jb1[k]);
#pragma unroll
  for (int j = 0; j < 6; ++j) {
    float acc = Jb2[j];
#pragma unroll
    for (int k = 0; k < 10; ++k) acc += Jw2[j*10 + k] * t[k];
    o[j] = acc;
  }
  psd_pack(o, Ji);
#pragma unroll
  for (int k = 0; k < 10; ++k) t[k] = fast_tanh(Jb1[k] - z[k]);
#pragma unroll
  for (int j = 0; j < 6; ++j) {
    float acc = Jb2[j];
#pragma unroll
    for (int k = 0; k < 10; ++k) acc += Jw2[j*10 + k] * t[k];
    o[j] = acc;
  }
  psd_pack(o, Pm);
#pragma unroll
  for (int i = 0; i < 6; ++i) Ji[i] = 0.5f * (Ji[i] + Pm[i]);

  // -------- V net gradient at q_k: one GEMM, both signs from z --------
  float diff[10];
  net_z(A0, A1, BV, stage, lane, z);
#pragma unroll
  for (int k = 0; k < 10; ++k) {
    float tp = fast_tanh(z[k] + Vb1[k]);
    float tm = fast_tanh(Vb1[k] - z[k]);
    diff[k] = 0.5f * (tm*tm - tp*tp) * Vw2[k];   // 0.5*((1-tp^2)-(1-tm^2))*w2
  }
  float dV[4];
#pragma unroll
  for (int i = 0; i < 4; ++i) {
    float acc = 0.0f;
#pragma unroll
    for (int k = 0; k < 10; ++k) acc += Vw1[k*4 + i] * diff[k];
    dV[i] = acc;
  }

  // -------- G net: g = 0.5*(mlp(q)+mlp(-q)), fk = 0.5*h*g*u --------
  float g3[3];
  net_z(A0, A1, BG, stage, lane, z);
  {
    float tp[10], tm[10];
#pragma unroll
    for (int k = 0; k < 10; ++k) {
      tp[k] = fast_tanh(z[k] + Gb1[k]);
      tm[k] = fast_tanh(Gb1[k] - z[k]);
    }
#pragma unroll
    for (int j = 0; j < 3; ++j) {
      float acc = 2.0f * Gb2[j];
#pragma unroll
      for (int k = 0; k < 10; ++k) acc += Gw2[j*10 + k] * (tp[k] + tm[k]);
      g3[j] = 0.5f * acc;
    }
  }
  float fk[3] = { 0.5f*HS*g3[0]*u, 0.5f*HS*g3[1]*u, 0.5f*HS*g3[2]*u };

  // -------- scalar tail: Jq, pk, RHS --------
  float Jq[6]; inv3sym(Ji, Jq);
  float pk[3]; symmv(Jq, wk, pk);
  pk[0]*=2.0f; pk[1]*=2.0f; pk[2]*=2.0f;
  float Hd[3]; hmat_mul(q, dV, Hd);
  float RHS[3];
#pragma unroll
  for (int i = 0; i < 3; ++i)
    RHS[i] = -0.25f*HS*(pk[i] + fk[i] - 0.5f*HS*Hd[i]);

  // -------- 5 Newton iterations, analytic fwd-mode jac + Cramer solve ------
  float xi[3] = {0.0f, 0.0f, 0.0f};
#pragma unroll
  for (int it = 0; it < 5; ++it) {
    float r[3], jc[9], dx[3];
    res_and_jac(xi, Jq, r, jc);
    r[0]-=RHS[0]; r[1]-=RHS[1]; r[2]-=RHS[2];
    solve3(jc, r, dx);
    xi[0]-=dx[0]; xi[1]-=dx[1]; xi[2]-=dx[2];
  }

  // -------- q_next = q_k * exp(xi) --------
  float dot = xi[0]*xi[0]+xi[1]*xi[1]+xi[2]*xi[2];
  float th  = sqrtf(fmaxf(dot, 1e-24f));
  float so  = __sinf(th) * rcpf(th);
  float qe[4] = { __cosf(th), xi[0]*so, xi[1]*so, xi[2]*so };
  float qn[4]; qmul(q, qe, qn);

  // -------- V gradient at q_next (2nd WMMA phase) --------
  qst[wid][lane][0]=qn[0]; qst[wid][lane][1]=qn[1];
  qst[wid][lane][2]=qn[2]; qst[wid][lane][3]=qn[3];
  __builtin_amdgcn_wave_barrier();
  v2f An0, An1;
  An0[0] = qst[wid][cm][2*half];     An0[1] = qst[wid][cm][2*half + 1];
  An1[0] = qst[wid][16+cm][2*half];  An1[1] = qst[wid][16+cm][2*half + 1];
  __builtin_amdgcn_wave_barrier();

  net_z(An0, An1, BV, stage, lane, z);
#pragma unroll
  for (int k = 0; k < 10; ++k) {
    float tp = fast_tanh(z[k] + Vb1[k]);
    float tm = fast_tanh(Vb1[k] - z[k]);
    diff[k] = 0.5f * (tm*tm - tp*tp) * Vw2[k];
  }
  float dVn[4];
#pragma unroll
  for (int i = 0; i < 4; ++i) {
    float acc = 0.0f;
#pragma unroll
    for (int k = 0; k < 10; ++k) acc += Vw1[k*4 + i] * diff[k];
    dVn[i] = acc;
  }

  // -------- p_next, w_next --------
  float qc[4] = { q[0], -q[1], -q[2], -q[3] };
  float qq[4]; qmul(qc, qn, qq);
  float qqv[3] = { qq[1], qq[2], qq[3] };
  float mv[3]; symmv(Jq, qqv, mv);
  float Hdn[3]; hmat_mul(qn, dVn, Hdn);
  float pn[3];
  pn[0] = (4.0f/HS)*(qq[0]*mv[0] - (qqv[1]*mv[2]-qqv[2]*mv[1])) - 0.5f*HS*Hdn[0] + fk[0];
  pn[1] = (4.0f/HS)*(qq[0]*mv[1] - (qqv[2]*mv[0]-qqv[0]*mv[2])) - 0.5f*HS*Hdn[1] + fk[1];
  pn[2] = (4.0f/HS)*(qq[0]*mv[2] - (qqv[0]*mv[1]-qqv[1]*mv[0])) - 0.5f*HS*Hdn[2] + fk[2];
  float wn[3]; symmv(Ji, pn, wn);
  wn[0]*=0.5f; wn[1]*=0.5f; wn[2]*=0.5f;

  // -------- write out (2x B128) --------
  float4 o0 = { qn[0], qn[1], qn[2], qn[3] };
  float4 o1 = { wn[0], wn[1], wn[2], u };
  float4* oe = reinterpret_cast<float4*>(out) + e * 2;
  oe[0] = o0;
  oe[1] = o1;
}

extern "C" void kernel_launch(void* const* d_in, const int* in_sizes, int n_in,
                              void* d_out, int out_size, void* d_ws, size_t ws_size,
                              hipStream_t stream) {
  const float* x   = (const float*)d_in[0];
  const float* Jw1 = (const float*)d_in[1];
  const float* Jb1 = (const float*)d_in[2];
  const float* Jw2 = (const float*)d_in[3];
  const float* Jb2 = (const float*)d_in[4];
  const float* Vw1 = (const float*)d_in[5];
  const float* Vb1 = (const float*)d_in[6];
  const float* Vw2 = (const float*)d_in[7];
  // d_in[8] = Vb2 (unused: only grad of V is needed)
  const float* Gw1 = (const float*)d_in[9];
  const float* Gb1 = (const float*)d_in[10];
  const float* Gw2 = (const float*)d_in[11];
  const float* Gb2 = (const float*)d_in[12];

  const int n = in_sizes[0] / 8;          // B = 1048576 -> 4096 blocks
  dim3 grid(n / 256), block(256);
  qvi_kernel<<<grid, block, 0, stream>>>(x, Jw1, Jb1, Jw2, Jb2,
                                         Vw1, Vb1, Vw2,
                                         Gw1, Gb1, Gw2, Gb2,
                                         (float*)d_out);
}